// CTCGreedyDecoder_23665269801320
// MI455X (gfx1250) — compile-verified
//
#include <hip/hip_runtime.h>
#include <stdint.h>

// ---------------- problem constants (reference: SEQ_LEN=2000, BATCH=32, ALPHABET=1024) ----
#define T_LEN   2000
#define BATCH   32
#define ALPHA   1024
#define ROWS    (T_LEN * BATCH)            // 64000 (t-major: row = t*BATCH + b)
#define ROWS_PER_TILE 8
#define NTILES  (ROWS / ROWS_PER_TILE)     // 8000
#define TILE_FLOATS (ROWS_PER_TILE * ALPHA) // 8192
#define TILE_BYTES  (TILE_FLOATS * 4)       // 32768

typedef uint32_t v4u __attribute__((ext_vector_type(4)));
typedef int      v8i __attribute__((ext_vector_type(8)));
typedef int      v4i __attribute__((ext_vector_type(4)));

#if defined(__has_builtin)
#if __has_builtin(__builtin_amdgcn_tensor_load_to_lds) && __has_builtin(__builtin_amdgcn_s_wait_tensorcnt)
#define USE_TDM 1
#endif
#endif

#ifdef USE_TDM
// Issue one TDM descriptor: contiguous 32KB (8192 fp32) tile  global -> LDS.
// D# packing per CDNA5 ISA.
// clang-23 lane builtin: (uint32x4 g0, int32x8 g1, int32x4 g2, int32x4 g3, int32x8, i32 cpol)
__device__ __forceinline__ void tdm_issue(const float* x, uint32_t tile, void* ldsPtr)
{
    uint64_t g  = (uint64_t)(uintptr_t)x + (uint64_t)tile * (uint64_t)TILE_BYTES;
    // Low 32 bits of a flat shared-aperture address ARE the LDS byte offset
    // (aperture lives in addr[63:32], ISA 10.2).
    uint32_t la = (uint32_t)(uintptr_t)ldsPtr;
    uint32_t lo = (uint32_t)g;
    uint32_t hi = (uint32_t)(g >> 32);

    v4u g0;
    g0.x = 1u;                                                        // count=1, user mode
    g0.y = (uint32_t)__builtin_amdgcn_readfirstlane((int)la);         // lds_addr [63:32]
    g0.z = (uint32_t)__builtin_amdgcn_readfirstlane((int)lo);         // global_addr[31:0]
    g0.w = ((uint32_t)__builtin_amdgcn_readfirstlane((int)hi) & 0x01FFFFFFu)
           | 0x80000000u;                                             // addr[56:32] | type=2

    v8i g1;
    g1[0] = 0x00020000;   // workgroup_mask=0, data_size=2 (4B), no barrier/iterate/pad
    g1[1] = 0x20000000;   // atomic_barrier_addr=0 ; tensor_dim0 = 8192 (low16<<16)
    g1[2] = 0x00010000;   // tensor_dim0 hi16 = 0 ; tensor_dim1 = 1
    g1[3] = 0x20000000;   // tensor_dim1 hi16 = 0 ; tile_dim0 = 8192
    g1[4] = 0x00000001;   // tile_dim1 = 1 ; tile_dim2 = 0
    g1[5] = 8192;         // tensor_dim0_stride (elements)
    g1[6] = 0;
    g1[7] = 0;

    v4i z4 = {0, 0, 0, 0};                  // groups 2/3 unused (<=2D tensor)
    v8i z8 = {0, 0, 0, 0, 0, 0, 0, 0};      // extra 6th-form operand, zero-filled
    __builtin_amdgcn_tensor_load_to_lds(g0, g1, z4, z4, z8, 0);
}
#endif

// ---------------- stage 1: per-(t,b) argmax over ALPHA=1024, written as ml[b][t] ----------
__global__ void __launch_bounds__(256) ctc_argmax_kernel(const float* __restrict__ x,
                                                         int* __restrict__ ml)
{
    const int lane = threadIdx.x & 31;
    const int wave = threadIdx.x >> 5;    // 8 waves, one row each per tile

#ifdef USE_TDM
    __shared__ __align__(16) float lds[2][ROWS_PER_TILE][ALPHA];   // 64 KB double buffer
    if (threadIdx.x < 32) {
        uint32_t first = blockIdx.x;
        if (first < NTILES) tdm_issue(x, first, &lds[0][0][0]);    // prologue: tile0 -> buf0
    }
#endif

    int it = 0;
    for (uint32_t tile = blockIdx.x; tile < NTILES; tile += gridDim.x, ++it) {
#ifdef USE_TDM
        const int cur = it & 1;
        __builtin_amdgcn_s_wait_tensorcnt(0);     // wave0: current tile landed (no-op for others)
        __syncthreads();                          // buf[cur] visible to all waves
        if (threadIdx.x < 32) {
            uint32_t nxt = tile + gridDim.x;
            if (nxt < NTILES) tdm_issue(x, nxt, &lds[cur ^ 1][0][0]);
        }
        const float4* rowp = (const float4*)&lds[cur][wave][0];
#else
        const float4* rowp = (const float4*)(x + (size_t)(tile * ROWS_PER_TILE + wave) * ALPHA);
#endif

        // Lane-local argmax over 8 float4s (indices ascend -> strict '>' keeps first max).
        float bv = -__builtin_huge_valf();
        int   bi = 0;
        #pragma unroll
        for (int c = 0; c < 8; ++c) {
            int    q  = c * 32 + lane;
            float4 v  = rowp[q];
            int    i0 = q * 4;
            if (v.x > bv) { bv = v.x; bi = i0;     }
            if (v.y > bv) { bv = v.y; bi = i0 + 1; }
            if (v.z > bv) { bv = v.z; bi = i0 + 2; }
            if (v.w > bv) { bv = v.w; bi = i0 + 3; }
        }
        // Cross-lane reduce; on ties keep the smaller index (jnp.argmax semantics).
        #pragma unroll
        for (int off = 16; off; off >>= 1) {
            float ov = __shfl_xor(bv, off, 32);
            int   oi = __shfl_xor(bi, off, 32);
            if (ov > bv || (ov == bv && oi < bi)) { bv = ov; bi = oi; }
        }
        if (lane == 0) {
            int row = (int)tile * ROWS_PER_TILE + wave;  // row = t*BATCH + b
            int b   = row & (BATCH - 1);
            int t   = row >> 5;                          // BATCH == 32
            ml[b * T_LEN + t] = bi;
        }
#ifdef USE_TDM
        __syncthreads();   // everyone done reading buf[cur] before it is refilled
#endif
    }
}

// ---------------- stage 2: per-batch dedup/compaction (one wave per batch row) ------------
// The ml row (8 KB) is staged into LDS with pipelined b128 loads so the serial
// ballot/scan chain only touches low-latency LDS; `prev` comes from shfl_up + a
// cross-chunk carry lane (no second load on the critical path).
__global__ void __launch_bounds__(32) ctc_compact_kernel(const int* __restrict__ ml,
                                                         const int* __restrict__ lengths,
                                                         int* __restrict__ out)
{
    __shared__ __align__(16) int s_ml[T_LEN];             // 8 KB

    const int b    = blockIdx.x;
    const int lane = threadIdx.x;
    const int len  = lengths[b];
    const int* row = ml  + b * T_LEN;
    int*       tok = out + b * T_LEN;

    // Stage the row into LDS: 500 int4s, 16 independent loads/lane (one latency).
    const int4* row4 = (const int4*)row;
    int4*       s4   = (int4*)s_ml;
    #pragma unroll 4
    for (int q = lane; q < T_LEN / 4; q += 32) s4[q] = row4[q];

    // Fill the output row with the pad value.
    for (int t = lane; t < T_LEN; t += 32) tok[t] = -1;

    int base  = 0;
    int carry = -1;                                        // prev of t==0 sentinel
    for (int t0 = 0; t0 < T_LEN; t0 += 32) {
        int  t    = t0 + lane;
        bool inb  = t < T_LEN;
        int  m    = inb ? s_ml[t] : 0;
        int  mp   = __shfl_up(m, 1, 32);
        if (lane == 0) mp = carry;
        int  last = __shfl(m, 31, 32);                     // raw ml of chunk's last frame

        bool keep = inb && (t < len) && (m != 0) && (m != mp);
        unsigned mask = (unsigned)__ballot(keep);          // wave32: low 32 bits
        if (keep) {
            int pos = base + __popc(mask & ((1u << lane) - 1u));
            tok[pos] = m;
        }
        base += __popc(mask);
        carry = last;
    }
    if (lane == 0) out[BATCH * T_LEN + b] = base;          // out_lengths[b]
}

// ---------------- launch ------------------------------------------------------------------
extern "C" void kernel_launch(void* const* d_in, const int* in_sizes, int n_in,
                              void* d_out, int out_size, void* d_ws, size_t ws_size,
                              hipStream_t stream)
{
    const float* x       = (const float*)d_in[0];   // (2000, 32, 1024) fp32
    const int*   lengths = (const int*)d_in[1];     // (32,) int32
    int*         out     = (int*)d_out;             // 64000 tokens + 32 lengths
    int*         ml      = (int*)d_ws;              // (32, 2000) int32 scratch = 256 KB

    // 2000 blocks x 4 tiles each: every block double-buffers 32KB TDM tiles through LDS.
    ctc_argmax_kernel<<<2000, 256, 0, stream>>>(x, ml);
    ctc_compact_kernel<<<BATCH, 32, 0, stream>>>(ml, lengths, out);
}